// ALiBi_61821759259207
// MI455X (gfx1250) — compile-verified
//
#include <hip/hip_runtime.h>

typedef __attribute__((ext_vector_type(2))) float v2f;
typedef __attribute__((ext_vector_type(4))) float v4f;
typedef __attribute__((ext_vector_type(8))) float v8f;

// Fixed by reference setup: B=2, H=16, S=2048
constexpr int H = 16;
constexpr int S = 2048;
constexpr int CHUNK = 256;                 // floats handled per wave per WMMA tile
constexpr int CHUNKS_PER_ROW = S / CHUNK;  // 8

// One wave per row of the [B,H,S,S] tensor. Each chunk of 256 contiguous
// floats is treated as the C operand of a 16x16 f32 WMMA tile; the ALiBi
// bias (a rank-2 function of (i,j)) is injected as A(16x4) x B(4x16).
__global__ __launch_bounds__(256) void alibi_wmma_kernel(
    const float* __restrict__ scores, float* __restrict__ out)
{
  const int wavesPerBlock = blockDim.x >> 5;
  const int wave = blockIdx.x * wavesPerBlock + (threadIdx.x >> 5);
  const int lane = threadIdx.x & 31;

  // row = ((b*H + h)*S + i)
  const int h = (wave >> 11) & (H - 1);   // (wave / S) % H, S=2048
  const int i = wave & (S - 1);           // query position
  const long long rowBase = (long long)wave * (long long)S;

  // ALiBi slope: 2^(-8*(h+1)/H)
  const float slope = exp2f(-8.0f * (float)(h + 1) / (float)H);

  // Slot -> memory mapping for C (two b128 loads per lane):
  //   element e = 4*lane + (v%4) + 128*(v/4)
  // With C layout (VGPR v holds M = v%8 [+8 for lanes>=16], N = lane%16):
  //   e = 4N + 64*(M/8) + (M%8)%4 + 128*((M%8)/4)
  // out = scores + slope*(j - i) = C + A x B with:
  //   A[M][0] = 1,            B[0][N] = slope*(j0 - i + 4N)
  //   A[M][1] = slope*eM(M),  B[1][N] = 1
  const bool lo = lane < 16;
  const int M = lane & 15;
  const int m = M & 7;
  const float eM = 64.0f * (float)(M >> 3)
                 + (float)(m & 3)
                 + 128.0f * (float)(m >> 2);

  v2f A;
  A.x = lo ? 1.0f : 0.0f;          // K=0 (lanes<16) / K=2 (lanes>=16)
  A.y = lo ? slope * eM : 0.0f;    // K=1 / K=3

  v2f Bm;
  Bm.y = lo ? 1.0f : 0.0f;         // B[1][N]=1, B[3][N]=0

  const float base_jmi = (float)(4 * M - i);  // 4N - i  (N == lane&15)

  const float* rp = scores + rowBase;
  float* wp = out + rowBase;

  #pragma unroll
  for (int c = 0; c < CHUNKS_PER_ROW; ++c) {
    const int cb = c * CHUNK;
    // B[0][N] = slope*(j0 - i + 4N); K=2 row (lanes>=16) stays zero
    Bm.x = lo ? slope * ((float)cb + base_jmi) : 0.0f;

    // Streaming 128-bit non-temporal loads: 2 x 512B per wave
    const v4f* lp = (const v4f*)(rp + cb) + lane;
    v4f d0 = __builtin_nontemporal_load(lp);
    v4f d1 = __builtin_nontemporal_load(lp + 32);

    v8f C;
    C[0] = d0.x; C[1] = d0.y; C[2] = d0.z; C[3] = d0.w;
    C[4] = d1.x; C[5] = d1.y; C[6] = d1.z; C[7] = d1.w;

    // out_tile = A x B + scores_tile   (single matrix op per 256 floats)
    v8f D = __builtin_amdgcn_wmma_f32_16x16x4_f32(
        false, A, false, Bm, (short)0, C, false, false);

    v4f o0, o1;
    o0.x = D[0]; o0.y = D[1]; o0.z = D[2]; o0.w = D[3];
    o1.x = D[4]; o1.y = D[5]; o1.z = D[6]; o1.w = D[7];

    v4f* op = (v4f*)(wp + cb) + lane;
    __builtin_nontemporal_store(o0, op);
    __builtin_nontemporal_store(o1, op + 32);
  }
}

extern "C" void kernel_launch(void* const* d_in, const int* in_sizes, int n_in,
                              void* d_out, int out_size, void* d_ws, size_t ws_size,
                              hipStream_t stream) {
  (void)in_sizes; (void)n_in; (void)d_ws; (void)ws_size;
  const float* scores = (const float*)d_in[0];
  // d_in[1] is `offset`; it cancels exactly in pos_diff, so it is unused.
  float* out = (float*)d_out;

  const int rows = out_size / S;             // B*H*S = 65536 waves
  const int block = 256;                     // 8 waves per block
  const int wavesPerBlock = block / 32;
  const int grid = rows / wavesPerBlock;     // 8192 blocks

  alibi_wmma_kernel<<<grid, block, 0, stream>>>(scores, out);
}